// ConvNet_78881369358604
// MI455X (gfx1250) — compile-verified
//
#include <hip/hip_runtime.h>

typedef __attribute__((ext_vector_type(2))) float v2f;
typedef __attribute__((ext_vector_type(4))) float v4f;
typedef __attribute__((ext_vector_type(8))) float v8f;

namespace {
constexpr int kB = 16;
constexpr int kV = 20000;
constexpr int kF = 64;      // FIN == COUT == 64
constexpr int kK = 16;      // neighbors
constexpr int kWaves = 4;   // waves per block (wave32)
constexpr int AST = 68;     // padded LDS tile row stride (floats): frag reads hit
                            // banks (4M + kb) % 64 -> all 64 banks once, conflict-free
}

// One wave32 per 16-row output tile. out = x*Wx + mean(nb)*Wn + b, exact f32 on
// V_WMMA_F32_16X16X4_F32. Weights pre-swizzled in LDS to per-lane fragment
// layout so every B fragment is a single conflict-free ds_load_b64.
__global__ __launch_bounds__(128) void gnn_wmma_f32(
    const float* __restrict__ x,
    const float* __restrict__ Wx,
    const float* __restrict__ Wn,
    const float* __restrict__ bias,
    const int*   __restrict__ neighbor,
    float* __restrict__ out)
{
  // Swizzled weight fragments: flat d = (((m*4+nt)*16+kk)*32 + lane)*2 + p
  // holds W[m][4*kk + 2*(lane>>4) + p][nt*16 + (lane&15)]   (32 KB)
  __shared__ __align__(16) float lds_wf[2 * 4 * 16 * 64];
  // Per-wave x-tile and mean-tile, padded stride (34 KB). Also reused as
  // (a) staging buffer for the weight swizzle, (b) output transpose scratch.
  __shared__ __align__(16) float lds_a[kWaves][2][16 * AST];
  __shared__ __align__(16) float lds_bias[kF];

  const int tid  = threadIdx.x;
  const int wave = tid >> 5;
  const int lane = tid & 31;

  // ---- Phase 0a: stage Wx|Wn linearly into scratch (coalesced b128) ----
  float* wtmp = &lds_a[0][0][0];                    // 8704 floats >= 8192
  for (int i = tid * 4; i < kF * kF; i += 128 * 4) {
    *(v4f*)&wtmp[i]           = *(const v4f*)&Wx[i];
    *(v4f*)&wtmp[kF * kF + i] = *(const v4f*)&Wn[i];
  }
  if (tid < kF) lds_bias[tid] = bias[tid];
  __syncthreads();

  // ---- Phase 0b: swizzle into per-lane fragment layout ----
  for (int d = tid * 2; d < 2 * 4 * 16 * 64; d += 128 * 2) {
    const int ln = (d >> 1) & 31;
    const int kk = (d >> 6) & 15;
    const int nt = (d >> 10) & 3;
    const int m  = (d >> 12) & 1;
    const int k0 = 4 * kk + 2 * (ln >> 4);
    const int sc = m * (kF * kF) + nt * 16 + (ln & 15);
    lds_wf[d]     = wtmp[sc + (k0    ) * kF];
    lds_wf[d + 1] = wtmp[sc + (k0 + 1) * kF];
  }
  __syncthreads();   // wtmp dead; lds_a now free for tiles

  const int  tile = blockIdx.x * kWaves + wave;     // 0..19999
  const long r0   = (long)tile * 16;                // first global row (b*V+v)
  const int  bb   = (int)(r0 / kV);
  const int  v0   = (int)(r0 % kV);                 // tiles never straddle a batch

  float* axt = &lds_a[wave][0][0];                  // x tile, stride AST
  float* amt = &lds_a[wave][1][0];                  // mean tile, stride AST

  // ---- Phase 1a: coalesced cooperative x-tile load (8 x global_load_b128) ----
  const float* xt = x + r0 * kF;
#pragma unroll
  for (int i = 0; i < 8; ++i) {
    const int t = i * 128 + lane * 4;               // flat idx in 16x64 tile
    v4f v = *(const v4f*)&xt[t];
    *(v4f*)&axt[(t >> 6) * AST + (t & 63)] = v;
  }

  // ---- Phase 1b: gather + mean. Lane owns features 2l, 2l+1: each neighbor
  // row is read as one 256B-contiguous wave access (pure L2 hits). ----
  const int f = 2 * lane;
  for (int row = 0; row < 16; ++row) {
    const int* nbp = neighbor + (long)(v0 + row) * kK;   // uniform -> s_load
    float ax = 0.f, ay = 0.f;
#pragma unroll
    for (int j = 0; j < kK; ++j) {
      const int   idx = nbp[j];                    // 0 => zero pad
      const int   re  = (idx > 0) ? (idx - 1) : 0;
      const float mk  = (idx > 0) ? 1.0f : 0.0f;
      v2f val = *(const v2f*)(x + ((long)bb * kV + re) * kF + f);
      ax += mk * val[0];
      ay += mk * val[1];
    }
    v2f mv; mv[0] = ax * 0.0625f; mv[1] = ay * 0.0625f;
    *(v2f*)&amt[row * AST + f] = mv;               // conflict-free b64 store
  }
  __syncthreads();

  // ---- Phase 2: WMMA k-loop. 8 x v_wmma_f32_16x16x4_f32 per kk step. ----
  const int M = lane & 15;
  const int h = lane >> 4;
  v8f acc[4] = {};
  for (int kk = 0; kk < 16; ++kk) {
    const int kb = 4 * kk + 2 * h;
    v2f a1 = *(const v2f*)&axt[M * AST + kb];      // conflict-free b64
    v2f a2 = *(const v2f*)&amt[M * AST + kb];
#pragma unroll
    for (int nt = 0; nt < 4; ++nt) {
      v2f b1 = *(const v2f*)&lds_wf[((nt     ) * 16 + kk) * 64 + lane * 2];
      v2f b2 = *(const v2f*)&lds_wf[((nt +  4) * 16 + kk) * 64 + lane * 2];
      acc[nt] = __builtin_amdgcn_wmma_f32_16x16x4_f32(
          false, a1, false, b1, (short)0, acc[nt], false, false);
      acc[nt] = __builtin_amdgcn_wmma_f32_16x16x4_f32(
          false, a2, false, b2, (short)0, acc[nt], false, false);
    }
  }

  // ---- Phase 3: bias + transpose through LDS, 16 coalesced 256B row stores ----
#pragma unroll
  for (int nt = 0; nt < 4; ++nt) {
    const float bv = lds_bias[nt * 16 + M];
#pragma unroll
    for (int r = 0; r < 8; ++r)
      axt[(r + 8 * h) * AST + nt * 16 + M] = acc[nt][r] + bv;  // conflict-free
  }
  __syncthreads();
  float* orow = out + r0 * kF;
#pragma unroll
  for (int rw = 0; rw < 16; ++rw) {
    v2f v = *(const v2f*)&axt[rw * AST + 2 * lane];
    *(v2f*)&orow[rw * kF + 2 * lane] = v;          // 32 lanes x 8B = full row
  }
}

extern "C" void kernel_launch(void* const* d_in, const int* in_sizes, int n_in,
                              void* d_out, int out_size, void* d_ws, size_t ws_size,
                              hipStream_t stream) {
  const float* x  = (const float*)d_in[0];
  const float* Wx = (const float*)d_in[1];
  const float* Wn = (const float*)d_in[2];
  const float* b  = (const float*)d_in[3];
  const int*   nb = (const int*)d_in[4];
  float* out = (float*)d_out;

  const int tiles  = kB * kV / 16;   // 20000 sixteen-row tiles
  const int blocks = tiles / kWaves; // 5000 blocks x 128 threads (4 waves)
  gnn_wmma_f32<<<blocks, 128, 0, stream>>>(x, Wx, Wn, b, nb, out);
}